// MotifGNN_8916352106710
// MI455X (gfx1250) — compile-verified
//
#include <hip/hip_runtime.h>
#include <math.h>

// Problem constants (fixed by the reference)
constexpr int kN  = 50000;
constexpr int kE  = 600000;
constexpr int kF  = 128;
constexpr int kHD = 256;
constexpr int kH  = 4;
constexpr int kC  = 64;
constexpr int kL  = 3;
constexpr int kB  = 64;
constexpr int kEp = kE + kN;   // edges + self loops = 650000
constexpr float kEps   = 1e-5f;
constexpr float kSlope = 0.2f;

typedef float v2f __attribute__((ext_vector_type(2)));
typedef float v8f __attribute__((ext_vector_type(8)));

// ---------------------------------------------------------------------------
// Generic fill
// ---------------------------------------------------------------------------
__global__ void fill_f32(float* p, float v, long long n) {
  long long i = (long long)blockIdx.x * blockDim.x + threadIdx.x;
  if (i < n) p[i] = v;
}

// ---------------------------------------------------------------------------
// f32 GEMM via V_WMMA_F32_16X16X4_F32.
// C[M x Nout] = A[M x K] * W[K x Nout] (+ optional bias broadcast over rows)
// grid = (Nout/64, M/16), block = 32 (one wave per 16x64 strip).
// x = N-strip varies fastest so the 4 blocks consuming the same 16-row A tile
// are launch-adjacent -> A tile stays cache-resident (A streamed once from HBM).
// All tile dims divide evenly in every call, so EXEC is all ones (WMMA req).
// ---------------------------------------------------------------------------
__global__ void gemm_wmma_f32(const float* __restrict__ A,
                              const float* __restrict__ Wm,
                              const float* __restrict__ bias,
                              float* __restrict__ Cout,
                              int M, int K, int Nout) {
  const int lane = threadIdx.x;
  const int lo = lane & 15;
  const int hi = lane >> 4;
  const int n0 = blockIdx.x * 64;
  const int m0 = blockIdx.y * 16;

  v8f acc0 = {}; v8f acc1 = {}; v8f acc2 = {}; v8f acc3 = {};
  const float* arow = A + (size_t)(m0 + lo) * K;

  for (int k = 0; k < K; k += 4) {
    const int ka = k + 2 * hi;
    v2f a;
    a.x = arow[ka];
    a.y = arow[ka + 1];
    const float* w0 = Wm + (size_t)ka * Nout + n0 + lo;
    const float* w1 = w0 + Nout;
    v2f b0; b0.x = w0[0];  b0.y = w1[0];
    v2f b1; b1.x = w0[16]; b1.y = w1[16];
    v2f b2; b2.x = w0[32]; b2.y = w1[32];
    v2f b3; b3.x = w0[48]; b3.y = w1[48];
    acc0 = __builtin_amdgcn_wmma_f32_16x16x4_f32(false, a, false, b0, (short)0, acc0, false, false);
    acc1 = __builtin_amdgcn_wmma_f32_16x16x4_f32(false, a, false, b1, (short)0, acc1, false, false);
    acc2 = __builtin_amdgcn_wmma_f32_16x16x4_f32(false, a, false, b2, (short)0, acc2, false, false);
    acc3 = __builtin_amdgcn_wmma_f32_16x16x4_f32(false, a, false, b3, (short)0, acc3, false, false);
  }

  float bv0 = 0.f, bv1 = 0.f, bv2 = 0.f, bv3 = 0.f;
  if (bias) {
    bv0 = bias[n0 + lo];
    bv1 = bias[n0 + lo + 16];
    bv2 = bias[n0 + lo + 32];
    bv3 = bias[n0 + lo + 48];
  }
#pragma unroll
  for (int r = 0; r < 8; ++r) {
    size_t o = (size_t)(m0 + r + 8 * hi) * Nout + n0 + lo;
    Cout[o]      = acc0[r] + bv0;
    Cout[o + 16] = acc1[r] + bv1;
    Cout[o + 32] = acc2[r] + bv2;
    Cout[o + 48] = acc3[r] + bv3;
  }
}

// ---------------------------------------------------------------------------
// LayerNorm + ReLU (+ optional pre-bias, + optional residual).
// One block (256 threads) per row; HD == 256.
//   out = (resid? resid : 0) + relu( LN(xin + pre_bias) * g + b )
// ---------------------------------------------------------------------------
__global__ void ln_relu(const float* __restrict__ xin,
                        const float* __restrict__ pre_bias,
                        const float* __restrict__ g,
                        const float* __restrict__ b,
                        const float* __restrict__ resid,
                        float* __restrict__ out) {
  __shared__ float red[kHD];
  const int row = blockIdx.x;
  const int t = threadIdx.x;
  const size_t o = (size_t)row * kHD + t;

  float v = xin[o];
  if (pre_bias) v += pre_bias[t];

  red[t] = v;
  __syncthreads();
  for (int s = kHD / 2; s > 0; s >>= 1) {
    if (t < s) red[t] += red[t + s];
    __syncthreads();
  }
  const float mu = red[0] * (1.0f / kHD);
  __syncthreads();

  const float d = v - mu;
  red[t] = d * d;
  __syncthreads();
  for (int s = kHD / 2; s > 0; s >>= 1) {
    if (t < s) red[t] += red[t + s];
    __syncthreads();
  }
  const float var = red[0] * (1.0f / kHD);

  float y = d * rsqrtf(var + kEps) * g[t] + b[t];
  y = fmaxf(y, 0.f);
  out[o] = resid ? (resid[o] + y) : y;
}

// ---------------------------------------------------------------------------
// Self-loop edge_attr: segment mean of edge_attr over dst.
// One wave per edge; lane handles 4 features via float4 load.
// ---------------------------------------------------------------------------
__global__ void loop_sum(const int* __restrict__ ei,
                         const float* __restrict__ ea,
                         float* __restrict__ sum,
                         float* __restrict__ cnt) {
  const int lane = threadIdx.x & 31;
  const long long e = (long long)blockIdx.x * (blockDim.x >> 5) + (threadIdx.x >> 5);
  if (e >= kE) return;
  const int d = ei[kE + e];
  const int f = lane * 4;
  const float4 v = *(const float4*)(ea + (size_t)e * kF + f);
  float* s = sum + (size_t)d * kF + f;
  atomicAdd(s + 0, v.x);
  atomicAdd(s + 1, v.y);
  atomicAdd(s + 2, v.z);
  atomicAdd(s + 3, v.w);
  if (lane == 0) atomicAdd(&cnt[d], 1.0f);
}

__global__ void loop_div(float* __restrict__ sum, const float* __restrict__ cnt) {
  long long idx = (long long)blockIdx.x * blockDim.x + threadIdx.x;
  if (idx >= (long long)kN * kF) return;
  const int n = (int)(idx >> 7);
  sum[idx] = sum[idx] / fmaxf(cnt[n], 1.0f);
}

// ---------------------------------------------------------------------------
// Monotonic f32 <-> u32 for atomic segment-max
// ---------------------------------------------------------------------------
__device__ __forceinline__ unsigned f32_ord(float x) {
  unsigned u = __float_as_uint(x);
  return (u & 0x80000000u) ? ~u : (u | 0x80000000u);
}
__device__ __forceinline__ float ord_f32(unsigned u) {
  return (u & 0x80000000u) ? __uint_as_float(u ^ 0x80000000u) : __uint_as_float(~u);
}

// ---------------------------------------------------------------------------
// Edge attention scoring: score[e,h] = dot(leaky_relu(xl[src]+xr[dst]+ee), att[h])
// One wave per edge: lane covers 8 contiguous channels (2x float4), so 8 lanes
// cover one head; per-head dot reduced with shfl_xor within the 8-lane group.
// ---------------------------------------------------------------------------
__global__ void edge_score(const int* __restrict__ ei,
                           const float* __restrict__ xl,
                           const float* __restrict__ xr,
                           const float* __restrict__ eemat,
                           const float* __restrict__ att,
                           float* __restrict__ sc,
                           unsigned* __restrict__ smax) {
  const int lane = threadIdx.x & 31;
  const long long e = (long long)blockIdx.x * (blockDim.x >> 5) + (threadIdx.x >> 5);
  if (e >= kEp) return;
  const int s = (e < kE) ? ei[e] : (int)(e - kE);
  const int d = (e < kE) ? ei[kE + e] : (int)(e - kE);

  const int cb = lane * 8;            // channel base; head = cb/64 = lane/8
  const float4* pl = (const float4*)(xl + (size_t)s * kHD + cb);
  const float4* pr = (const float4*)(xr + (size_t)d * kHD + cb);
  const float4* pe = (const float4*)(eemat + (size_t)e * kHD + cb);
  const float4* pa = (const float4*)(att + cb);

  float acc = 0.f;
#pragma unroll
  for (int q = 0; q < 2; ++q) {
    const float4 l = pl[q], r = pr[q], ev = pe[q], a = pa[q];
    float m;
    m = l.x + r.x + ev.x; m = (m > 0.f) ? m : kSlope * m; acc += m * a.x;
    m = l.y + r.y + ev.y; m = (m > 0.f) ? m : kSlope * m; acc += m * a.y;
    m = l.z + r.z + ev.z; m = (m > 0.f) ? m : kSlope * m; acc += m * a.z;
    m = l.w + r.w + ev.w; m = (m > 0.f) ? m : kSlope * m; acc += m * a.w;
  }
  // reduce the 8 lanes of each head group
  acc += __shfl_xor(acc, 1, 32);
  acc += __shfl_xor(acc, 2, 32);
  acc += __shfl_xor(acc, 4, 32);
  if ((lane & 7) == 0) {
    const int h = lane >> 3;
    sc[e * kH + h] = acc;
    atomicMax(&smax[(size_t)d * kH + h], f32_ord(acc));
  }
}

// a = exp(score - smax[dst]); denom[dst] += a   (thread per (edge, head))
__global__ void edge_exp(const int* __restrict__ ei,
                         float* __restrict__ sc,
                         const unsigned* __restrict__ smax,
                         float* __restrict__ denom) {
  long long idx = (long long)blockIdx.x * blockDim.x + threadIdx.x;
  if (idx >= (long long)kEp * kH) return;
  const int e = (int)(idx >> 2);
  const int h = (int)(idx & (kH - 1));
  const int d = (e < kE) ? ei[kE + e] : (e - kE);
  const float mx = ord_f32(smax[(size_t)d * kH + h]);
  const float a = expf(sc[idx] - mx);
  sc[idx] = a;
  atomicAdd(&denom[(size_t)d * kH + h], a);
}

// hnew[dst, ch] += (a/denom[dst,h]) * xl[src, ch]
// One wave per edge; lane handles 8 channels via 2x float4 gathers.
__global__ void edge_aggregate(const int* __restrict__ ei,
                               const float* __restrict__ sc,
                               const float* __restrict__ denom,
                               const float* __restrict__ xl,
                               float* __restrict__ hnew) {
  const int lane = threadIdx.x & 31;
  const long long e = (long long)blockIdx.x * (blockDim.x >> 5) + (threadIdx.x >> 5);
  if (e >= kEp) return;
  const int s = (e < kE) ? ei[e] : (int)(e - kE);
  const int d = (e < kE) ? ei[kE + e] : (int)(e - kE);

  const int cb = lane * 8;
  const int h = lane >> 3;
  const float alpha = sc[e * kH + h] / denom[(size_t)d * kH + h];

  const float4* px = (const float4*)(xl + (size_t)s * kHD + cb);
  float* po = hnew + (size_t)d * kHD + cb;
#pragma unroll
  for (int q = 0; q < 2; ++q) {
    const float4 v = px[q];
    atomicAdd(po + 4 * q + 0, alpha * v.x);
    atomicAdd(po + 4 * q + 1, alpha * v.y);
    atomicAdd(po + 4 * q + 2, alpha * v.z);
    atomicAdd(po + 4 * q + 3, alpha * v.w);
  }
}

// ---------------------------------------------------------------------------
// Global mean pool: one wave per node; lane handles 8 channels.
// ---------------------------------------------------------------------------
__global__ void pool_acc(const int* __restrict__ batch,
                         const float* __restrict__ h,
                         float* __restrict__ out,
                         float* __restrict__ pcnt) {
  const int lane = threadIdx.x & 31;
  const long long n = (long long)blockIdx.x * (blockDim.x >> 5) + (threadIdx.x >> 5);
  if (n >= kN) return;
  const int g = batch[n];
  const int cb = lane * 8;
  const float4* ph = (const float4*)(h + (size_t)n * kHD + cb);
  float* po = out + (size_t)g * kHD + cb;
#pragma unroll
  for (int q = 0; q < 2; ++q) {
    const float4 v = ph[q];
    atomicAdd(po + 4 * q + 0, v.x);
    atomicAdd(po + 4 * q + 1, v.y);
    atomicAdd(po + 4 * q + 2, v.z);
    atomicAdd(po + 4 * q + 3, v.w);
  }
  if (lane == 0) atomicAdd(&pcnt[g], 1.0f);
}

__global__ void pool_div(float* __restrict__ out, const float* __restrict__ pcnt) {
  long long idx = (long long)blockIdx.x * blockDim.x + threadIdx.x;
  if (idx >= (long long)kB * kHD) return;
  out[idx] = out[idx] / fmaxf(pcnt[idx >> 8], 1.0f);
}

// ---------------------------------------------------------------------------
// Host-side orchestration
// ---------------------------------------------------------------------------
static inline dim3 g1(long long n) { return dim3((unsigned)((n + 255) / 256)); }
// grid for wave-per-item kernels with 256-thread blocks (8 waves/block)
static inline dim3 gw(long long items) { return dim3((unsigned)((items + 7) / 8)); }

extern "C" void kernel_launch(void* const* d_in, const int* in_sizes, int n_in,
                              void* d_out, int out_size, void* d_ws, size_t ws_size,
                              hipStream_t stream) {
  (void)in_sizes; (void)n_in; (void)out_size; (void)ws_size;

  const float* x      = (const float*)d_in[0];
  const int*   ei     = (const int*)d_in[1];     // [2,E]: src = ei[0:E), dst = ei[E:2E)
  const float* ea     = (const float*)d_in[2];   // [E,F]
  const int*   batch  = (const int*)d_in[3];
  const float* enc_w  = (const float*)d_in[5];
  const float* enc_b  = (const float*)d_in[6];
  const float* enc_g  = (const float*)d_in[7];
  const float* enc_bt = (const float*)d_in[8];
  const float* wl     = (const float*)d_in[9];
  const float* bl     = (const float*)d_in[10];
  const float* wr     = (const float*)d_in[11];
  const float* br     = (const float*)d_in[12];
  const float* we     = (const float*)d_in[13];
  const float* att    = (const float*)d_in[14];
  const float* bias   = (const float*)d_in[15];
  const float* lg     = (const float*)d_in[16];
  const float* lb     = (const float*)d_in[17];
  float* out = (float*)d_out;

  // Workspace carving (floats)
  float* Wf = (float*)d_ws;
  size_t off = 0;
  float* h     = Wf + off; off += (size_t)kN * kHD;
  float* hnew  = Wf + off; off += (size_t)kN * kHD;
  float* xlb   = Wf + off; off += (size_t)kN * kHD;
  float* xrb   = Wf + off; off += (size_t)kN * kHD;
  float* lattr = Wf + off; off += (size_t)kN * kF;   // loop attr (sums, then mean)
  float* lcnt  = Wf + off; off += (size_t)kN;
  float* eemat = Wf + off; off += (size_t)kEp * kHD; // edge features per layer
  float* sca   = Wf + off; off += (size_t)kEp * kH;  // scores, then exp(score-max)
  unsigned* smax = (unsigned*)(Wf + off); off += (size_t)kN * kH;
  float* denom = Wf + off; off += (size_t)kN * kH;
  float* pcnt  = Wf + off; off += (size_t)kB;

  const dim3 wave(32);
  const dim3 t256(256);

  // ---- encoder: h = relu(LN(x @ enc_w + enc_b)) ----
  gemm_wmma_f32<<<dim3(kHD / 64, kN / 16), wave, 0, stream>>>(x, enc_w, enc_b, hnew, kN, kF, kHD);
  ln_relu<<<dim3(kN), t256, 0, stream>>>(hnew, nullptr, enc_g, enc_bt, nullptr, h);

  // ---- self-loop edge attr (segment mean over dst) ----
  fill_f32<<<g1((long long)kN * kF), t256, 0, stream>>>(lattr, 0.f, (long long)kN * kF);
  fill_f32<<<g1(kN), t256, 0, stream>>>(lcnt, 0.f, kN);
  loop_sum<<<gw(kE), t256, 0, stream>>>(ei, ea, lattr, lcnt);
  loop_div<<<g1((long long)kN * kF), t256, 0, stream>>>(lattr, lcnt);

  // ---- GATv2 layers ----
  for (int l = 0; l < kL; ++l) {
    const float* wl_l = wl + (size_t)l * kHD * kHD;
    const float* wr_l = wr + (size_t)l * kHD * kHD;
    const float* we_l = we + (size_t)l * kF * kHD;

    gemm_wmma_f32<<<dim3(kHD / 64, kN / 16), wave, 0, stream>>>(h, wl_l, bl + l * kHD, xlb, kN, kHD, kHD);
    gemm_wmma_f32<<<dim3(kHD / 64, kN / 16), wave, 0, stream>>>(h, wr_l, br + l * kHD, xrb, kN, kHD, kHD);
    // ee for real edges and for self-loop rows
    gemm_wmma_f32<<<dim3(kHD / 64, kE / 16), wave, 0, stream>>>(ea, we_l, nullptr, eemat, kE, kF, kHD);
    gemm_wmma_f32<<<dim3(kHD / 64, kN / 16), wave, 0, stream>>>(lattr, we_l, nullptr,
                                                                eemat + (size_t)kE * kHD, kN, kF, kHD);

    fill_f32<<<g1((long long)kN * kH), t256, 0, stream>>>((float*)smax, 0.f, (long long)kN * kH);
    fill_f32<<<g1((long long)kN * kH), t256, 0, stream>>>(denom, 0.f, (long long)kN * kH);

    edge_score<<<gw(kEp), t256, 0, stream>>>(ei, xlb, xrb, eemat,
                                             att + (size_t)l * kH * kC, sca, smax);
    edge_exp<<<g1((long long)kEp * kH), t256, 0, stream>>>(ei, sca, smax, denom);

    fill_f32<<<g1((long long)kN * kHD), t256, 0, stream>>>(hnew, 0.f, (long long)kN * kHD);
    edge_aggregate<<<gw(kEp), t256, 0, stream>>>(ei, sca, denom, xlb, hnew);

    // h = h + relu(LN(hnew + bias_l))
    ln_relu<<<dim3(kN), t256, 0, stream>>>(hnew, bias + l * kHD, lg + l * kHD, lb + l * kHD, h, h);
  }

  // ---- global mean pool ----
  fill_f32<<<g1((long long)kB * kHD), t256, 0, stream>>>(out, 0.f, (long long)kB * kHD);
  fill_f32<<<g1(kB), t256, 0, stream>>>(pcnt, 0.f, kB);
  pool_acc<<<gw(kN), t256, 0, stream>>>(batch, h, out, pcnt);
  pool_div<<<g1((long long)kB * kHD), t256, 0, stream>>>(out, pcnt);
}